// ProfitFocusedLoss_26396869001422
// MI455X (gfx1250) — compile-verified
//
#include <hip/hip_runtime.h>
#include <hip/hip_bf16.h>

// ProfitFocusedLoss for MI455X (gfx1250, wave32).
// Memory-bound streaming reduction (~112 MB read once -> one scalar).
// Roofline: 112 MB / 23.3 TB/s ~= 5 us floor; all loads are b128 (512 B per
// wave per stream) and transcendentals use the HW v_exp/v_log/v_rcp paths so
// VALU stays under the memory floor.
// Wave-level reduction uses V_WMMA_F32_16X16X4_F32 with an all-ones B matrix:
//   D[m][n] = sum_k A[m][k]  ->  row-sum folds lane m with lane m+16.

typedef float v2f __attribute__((ext_vector_type(2)));
typedef float v8f __attribute__((ext_vector_type(8)));

#define ALPHA_F 0.7f
#define BLOCK   256
#define VEC     4      // elements per thread per iteration (float4 loads)
#define GRAIN   4      // iterations per thread -> 4096 elems/block, 1024 blocks at N=4M

// Reduce one float across the 32 lanes of a wave using WMMA.
// A-matrix (16x4 f32, 2 VGPRs/lane): lane m (<16) holds A[m][0],A[m][1];
// lane m+16 holds A[m][2],A[m][3]. Partial goes in slot 0, zero in slot 1,
// B = all ones => rowsum[m] = partial[m] + partial[m+16].
// D (C/D layout): lanes 0-15 VGPR r = rowsum[r], lanes 16-31 VGPR r = rowsum[r+8].
// Summing d[0..7] gives the half-sums; one xor-16 shuffle completes it.
__device__ __forceinline__ float wave_reduce_wmma(float partial) {
    v2f a; a[0] = partial; a[1] = 0.0f;
    v2f b; b[0] = 1.0f;    b[1] = 1.0f;   // all-ones B: layout-independent
    v8f c = {};
    v8f d = __builtin_amdgcn_wmma_f32_16x16x4_f32(
        /*neg_a=*/false, a, /*neg_b=*/false, b,
        /*c_mod=*/(short)0, c, /*reuse_a=*/false, /*reuse_b=*/false);
    float s = d[0] + d[1] + d[2] + d[3] + d[4] + d[5] + d[6] + d[7];
    s += __shfl_xor(s, 16, 32);           // combine rows 0..7 with rows 8..15
    return s;                             // every lane holds the wave total
}

// Block-level reduction of three running sums. All threads are active here
// (bounds checks live inside the accumulation loops), so EXEC is all ones
// as WMMA requires.
__device__ __forceinline__ void block_reduce3(float& s0, float& s1, float& s2) {
    __shared__ float lds[BLOCK / 32][3];
    const int wave = threadIdx.x >> 5;
    const int lane = threadIdx.x & 31;
    s0 = wave_reduce_wmma(s0);
    s1 = wave_reduce_wmma(s1);
    s2 = wave_reduce_wmma(s2);
    if (lane == 0) { lds[wave][0] = s0; lds[wave][1] = s1; lds[wave][2] = s2; }
    __syncthreads();
    if (threadIdx.x == 0) {
        float a = 0.0f, b = 0.0f, c = 0.0f;
        for (int w = 0; w < BLOCK / 32; ++w) {
            a += lds[w][0]; b += lds[w][1]; c += lds[w][2];
        }
        s0 = a; s1 = b; s2 = c;
    }
}

// Branch-free per-element accumulation.
__device__ __forceinline__ void accum_elem(float predp, float p,
                                           float x0, float x1, float x2,
                                           int t, float dfwd, float dbwd,
                                           float& sq, float& ce, float& pw) {
    // price MSE
    const float d = predp - p;
    sq += d * d;
    // stable log-softmax cross-entropy over 3 logits (HW exp/log)
    const float m   = fmaxf(x0, fmaxf(x1, x2));
    const float s   = __expf(x0 - m) + __expf(x1 - m) + __expf(x2 - m);
    const float lse = m + __logf(s);
    const float xt  = (t == 0) ? x0 : ((t == 1) ? x1 : x2);
    ce += lse - xt;
    // argmax (first max, matching jnp.argmax) -> profit select, all VALU
    int amax = 0; float best = x0;
    if (x1 > best) { best = x1; amax = 1; }
    if (x2 > best) { best = x2; amax = 2; }
    const float profit = (amax == 0) ? dfwd : ((amax == 2) ? dbwd : 0.0f);
    pw += __builtin_amdgcn_rcpf(1.0f + __expf(-10.0f * profit));
}

__global__ void __launch_bounds__(BLOCK)
profit_loss_partial(const float* __restrict__ pred_prices,
                    const float* __restrict__ pred_actions,
                    const float* __restrict__ tgt_prices,
                    const int*   __restrict__ tgt_actions,
                    float*       __restrict__ partials,   // [numBlocks][3]
                    int n) {
    const int base = blockIdx.x * (BLOCK * GRAIN * VEC);
    float sq = 0.0f, ce = 0.0f, pw = 0.0f;

    #pragma unroll
    for (int it = 0; it < GRAIN; ++it) {
        const int i = base + (it * BLOCK + threadIdx.x) * VEC;   // first of 4 elems
        if (i + VEC <= n) {
            // Prefetch the next tile of the widest stream (speculative, safe).
            __builtin_prefetch(&pred_actions[3 * (i + BLOCK * VEC)], 0, 0);

            // b128 loads: wave reads 512 B contiguous per stream.
            const float4 pp4 = *reinterpret_cast<const float4*>(pred_prices + i);
            const float4 tp4 = *reinterpret_cast<const float4*>(tgt_prices  + i);
            const int4   ta4 = *reinterpret_cast<const int4*>  (tgt_actions + i);
            const float4 a0  = *reinterpret_cast<const float4*>(pred_actions + 3 * i);
            const float4 a1  = *reinterpret_cast<const float4*>(pred_actions + 3 * i + 4);
            const float4 a2  = *reinterpret_cast<const float4*>(pred_actions + 3 * i + 8);

            // Boundary neighbors: unconditional clamped loads (cache hits).
            const float pm = tgt_prices[(i > 0) ? (i - 1) : 0];
            const float pn = tgt_prices[(i + VEC < n) ? (i + VEC) : (n - 1)];

            const float p[4]  = { tp4.x, tp4.y, tp4.z, tp4.w };
            // diff_fwd[j] = p[j+1]-p[j] (0 at global n-1); diff_bwd[j] = p[j]-p[j-1] (0 at 0)
            const float dfwd[4] = {
                tp4.y - tp4.x, tp4.z - tp4.y, tp4.w - tp4.z,
                (i + 3 == n - 1) ? 0.0f : (pn - tp4.w) };
            const float dbwd[4] = {
                (i == 0) ? 0.0f : (tp4.x - pm),
                tp4.y - tp4.x, tp4.z - tp4.y, tp4.w - tp4.z };

            accum_elem(pp4.x, p[0], a0.x, a0.y, a0.z, ta4.x, dfwd[0], dbwd[0], sq, ce, pw);
            accum_elem(pp4.y, p[1], a0.w, a1.x, a1.y, ta4.y, dfwd[1], dbwd[1], sq, ce, pw);
            accum_elem(pp4.z, p[2], a1.z, a1.w, a2.x, ta4.z, dfwd[2], dbwd[2], sq, ce, pw);
            accum_elem(pp4.w, p[3], a2.y, a2.z, a2.w, ta4.w, dfwd[3], dbwd[3], sq, ce, pw);
        } else {
            // Rare tail: guarded scalar path (never taken when VEC | n).
            for (int j = 0; j < VEC; ++j) {
                const int g = i + j;
                if (g < n) {
                    const float tp   = tgt_prices[g];
                    const float fw   = (g + 1 < n) ? (tgt_prices[g + 1] - tp) : 0.0f;
                    const float bw   = (g > 0)     ? (tp - tgt_prices[g - 1]) : 0.0f;
                    accum_elem(pred_prices[g], tp,
                               pred_actions[3 * g], pred_actions[3 * g + 1],
                               pred_actions[3 * g + 2], tgt_actions[g],
                               fw, bw, sq, ce, pw);
                }
            }
        }
    }

    block_reduce3(sq, ce, pw);
    if (threadIdx.x == 0) {
        partials[3 * blockIdx.x + 0] = sq;
        partials[3 * blockIdx.x + 1] = ce;
        partials[3 * blockIdx.x + 2] = pw;
    }
}

__global__ void __launch_bounds__(BLOCK)
profit_loss_final(const float* __restrict__ partials, int num_parts,
                  float* __restrict__ out, float inv_n) {
    float sq = 0.0f, ce = 0.0f, pw = 0.0f;
    for (int i = threadIdx.x; i < num_parts; i += BLOCK) {
        sq += partials[3 * i + 0];
        ce += partials[3 * i + 1];
        pw += partials[3 * i + 2];
    }
    block_reduce3(sq, ce, pw);
    if (threadIdx.x == 0) {
        const float price_loss  = sq * inv_n;
        const float action_loss = ce * inv_n;
        const float mean_pw     = pw * inv_n;
        out[0] = ALPHA_F * (action_loss * mean_pw) + (1.0f - ALPHA_F) * price_loss;
    }
}

extern "C" void kernel_launch(void* const* d_in, const int* in_sizes, int n_in,
                              void* d_out, int out_size, void* d_ws, size_t ws_size,
                              hipStream_t stream) {
    const float* pred_prices  = (const float*)d_in[0];
    const float* pred_actions = (const float*)d_in[1];
    const float* tgt_prices   = (const float*)d_in[2];
    const int*   tgt_actions  = (const int*)  d_in[3];
    // d_in[4] (prices) is unused by the reference.

    const int n  = in_sizes[0];
    const int elems_per_block = BLOCK * GRAIN * VEC;            // 4096
    const int nb = (n + elems_per_block - 1) / elems_per_block; // 1024 for N=4M

    float* partials = (float*)d_ws;   // nb*3 floats — fully overwritten each call
    float* out      = (float*)d_out;

    profit_loss_partial<<<nb, BLOCK, 0, stream>>>(
        pred_prices, pred_actions, tgt_prices, tgt_actions, partials, n);
    profit_loss_final<<<1, BLOCK, 0, stream>>>(
        partials, nb, out, 1.0f / (float)n);
}